// SymmetryLoss_89404039233675
// MI455X (gfx1250) — compile-verified
//
#include <hip/hip_runtime.h>
#include <math.h>

typedef __attribute__((ext_vector_type(2))) float v2f;
typedef __attribute__((ext_vector_type(8))) float v8f;

#define B_SZ    256
#define N_PTS   4096
#define GSZ     32
#define G3      32768
#define SPLIT   2
#define THREADS 256          // 8 waves of 32
#define WAVES   (THREADS / 32)
#define PTS_PER_BLOCK (N_PTS / SPLIT)        // 2048
#define CHUNKS        (PTS_PER_BLOCK / 16)   // 128 chunks of 16 points

// ---------------------------------------------------------------------------
// Init kernel: zero the 6 outputs and add the rotation theta penalty
// rot_loss[a] += mean_b( 1 / (theta_b + 1e-12) )
// ---------------------------------------------------------------------------
__global__ __launch_bounds__(768) void symmetry_init(const float* __restrict__ axes,
                                                     float* __restrict__ out) {
    int tid = threadIdx.x;            // 768 threads: a = tid/256, b = tid%256
    if (tid < 6) out[tid] = 0.0f;
    __syncthreads();
    int a = tid / B_SZ;
    int b = tid % B_SZ;
    float q0 = axes[((size_t)a * B_SZ + b) * 4 + 0];
    float theta = acosf(q0) * 2.0f * (180.0f / 3.14159265358979323846f);
    if (theta > 180.0f) theta = 360.0f - theta;
    atomicAdd(&out[3 + a], (1.0f / (theta + 1e-12f)) * (1.0f / (float)B_SZ));
}

// ---------------------------------------------------------------------------
// Main kernel: one block per (batch b, N-slice). All 6 affine transforms of 16
// points are computed with two V_WMMA_F32_16X16X4_F32 ops (18x4 stack), then
// each lane gathers/accumulates 3 transforms. Voxel mask is a 4KB LDS bitmask
// (ballot-packed from b128 loads); a set bit skips the 12B closest gather.
// ---------------------------------------------------------------------------
__global__ __launch_bounds__(THREADS) void symmetry_main(
    const float* __restrict__ planes,   // (3, B, 4)
    const float* __restrict__ axes,     // (3, B, 4)
    const float* __restrict__ samples,  // (B, N, 3)
    const float* __restrict__ closest,  // (B, G3, 3)
    const float* __restrict__ voxel,    // (B, G3)
    float* __restrict__ out)            // 6 floats
{
    __shared__ uint32_t maskbits[G3 / 32];   // 4 KB voxel-occupancy bitmask
    __shared__ float    M[18][4];            // stacked affine transforms
    __shared__ float    accum[6];

    const int b    = blockIdx.x;
    const int tid  = threadIdx.x;
    const int lane = tid & 31;
    const int wave = tid >> 5;
    const int r    = lane & 15;              // point index within chunk / A row
    const bool upper = (lane >= 16);

    if (tid < 6) accum[tid] = 0.0f;

    // --- Build the 18x4 transform matrix (rows 0-8: reflections, 9-17: rotations)
    if (tid < 18) {
        int t = tid / 3, c = tid % 3;
        float r0, r1, r2, r3;
        if (t < 3) {
            const float* pl = planes + ((size_t)t * B_SZ + b) * 4;
            float n0 = pl[0], n1 = pl[1], n2 = pl[2], d = pl[3];
            float s  = 2.0f / (n0 * n0 + n1 * n1 + n2 * n2 + 1e-12f);
            float nc = (c == 0) ? n0 : (c == 1) ? n1 : n2;
            r0 = ((c == 0) ? 1.0f : 0.0f) - s * nc * n0;
            r1 = ((c == 1) ? 1.0f : 0.0f) - s * nc * n1;
            r2 = ((c == 2) ? 1.0f : 0.0f) - s * nc * n2;
            r3 = -s * nc * d;
        } else {
            const float* q = axes + ((size_t)(t - 3) * B_SZ + b) * 4;
            float w = q[0], x = q[1], y = q[2], z = q[3];
            float k = w * w - (x * x + y * y + z * z);
            // q p conj(q):  (w^2-|v|^2) I + 2 v v^T + 2 w [v]x  (exact, any |q|)
            float m00 = k + 2*x*x,      m01 = 2*x*y - 2*w*z, m02 = 2*x*z + 2*w*y;
            float m10 = 2*x*y + 2*w*z,  m11 = k + 2*y*y,     m12 = 2*y*z - 2*w*x;
            float m20 = 2*x*z - 2*w*y,  m21 = 2*y*z + 2*w*x, m22 = k + 2*z*z;
            if (c == 0)      { r0 = m00; r1 = m01; r2 = m02; }
            else if (c == 1) { r0 = m10; r1 = m11; r2 = m12; }
            else             { r0 = m20; r1 = m21; r2 = m22; }
            r3 = 0.0f;
        }
        M[tid][0] = r0; M[tid][1] = r1; M[tid][2] = r2; M[tid][3] = r3;
    }

    // --- Stage voxel mask bits with b128 loads; permuted bit layout:
    //     element e -> word (e>>7)*4 + (e&3), bit (e>>2)&31
    const float* vox = voxel + (size_t)b * G3;
    for (int blk = wave; blk < G3 / 128; blk += WAVES) {
        float4 v = reinterpret_cast<const float4*>(vox)[blk * 32 + lane];
        uint32_t b0 = (uint32_t)__ballot(v.x != 0.0f);
        uint32_t b1 = (uint32_t)__ballot(v.y != 0.0f);
        uint32_t b2 = (uint32_t)__ballot(v.z != 0.0f);
        uint32_t b3 = (uint32_t)__ballot(v.w != 0.0f);
        if (lane == 0) {
            uint4 w4; w4.x = b0; w4.y = b1; w4.z = b2; w4.w = b3;
            reinterpret_cast<uint4*>(maskbits)[blk] = w4;
        }
    }
    __syncthreads();

    // --- Per-lane WMMA A operands (A is 16x4 f32: lanes<16 hold cols 0,1; >=16 cols 2,3)
    v2f a1, a2;
    a1.x = upper ? M[r][2] : M[r][0];
    a1.y = upper ? M[r][3] : M[r][1];
    a2.x = (r < 2) ? (upper ? M[16 + r][2] : M[16 + r][0]) : 0.0f;
    a2.y = (r < 2) ? (upper ? M[16 + r][3] : M[16 + r][1]) : 0.0f;

    const float* smp = samples + (size_t)b * N_PTS * 3;
    const float* cls = closest + (size_t)b * (size_t)G3 * 3;

    float acc0 = 0.0f, acc1 = 0.0f, acc2 = 0.0f;

    auto probe = [&](float tx, float ty, float tz, float& acc) {
        int ix = min(GSZ - 1, max(0, (int)floorf((tx + 0.5f) * (float)GSZ)));
        int iy = min(GSZ - 1, max(0, (int)floorf((ty + 0.5f) * (float)GSZ)));
        int iz = min(GSZ - 1, max(0, (int)floorf((tz + 0.5f) * (float)GSZ)));
        int flat = ix * (GSZ * GSZ) + iy * GSZ + iz;
        uint32_t bits = maskbits[((flat >> 7) << 2) | (flat & 3)];
        if (((bits >> ((flat >> 2) & 31)) & 1u) == 0u) {
            const float* cp = cls + (size_t)flat * 3;
            float dx = tx - cp[0], dy = ty - cp[1], dz = tz - cp[2];
            acc += dx * dx + dy * dy + dz * dz;
        }
    };

    const int lane_sel = upper ? 2 : 0;
    const int lane_adj = upper ? -1 : 1;

    for (int ch = wave; ch < CHUNKS; ch += WAVES) {   // uniform per wave: EXEC all-ones at WMMA
        const int n0 = blockIdx.y * PTS_PER_BLOCK + ch * 16;
        const int pt = n0 + r;

        // Speculative prefetch of next chunk's sample row (global_prefetch_b8)
        const int nch = (ch + WAVES < CHUNKS) ? (ch + WAVES) : ch;
        const int npt = blockIdx.y * PTS_PER_BLOCK + nch * 16 + r;
        __builtin_prefetch(&smp[npt * 3], 0, 1);

        // B is 4x16 homogeneous points: lanes<16 hold (x,y), lanes>=16 hold (z,1).
        // Branchless, always in-bounds (upper lanes re-read y and discard it).
        const int off0 = pt * 3 + lane_sel;
        float v0 = smp[off0];
        float v1 = smp[off0 + lane_adj];
        v2f bop; bop.x = v0; bop.y = upper ? 1.0f : v1;

        v8f c0 = {};
        v8f d1 = __builtin_amdgcn_wmma_f32_16x16x4_f32(false, a1, false, bop,
                                                       (short)0, c0, false, false);
        v8f d2 = __builtin_amdgcn_wmma_f32_16x16x4_f32(false, a2, false, bop,
                                                       (short)0, c0, false, false);

        // D rows: lanes<16 own rows 0-7 (t0,t1,t2.xy), lanes>=16 own rows 8-15
        // (t2.z, t3, t4, t5.x). Exchange t2.z down, t5.y/z up via xor-16.
        float send1 = upper ? d1[0] : d2[0];
        float send2 = upper ? 0.0f  : d2[1];
        float recv1 = __shfl_xor(send1, 16, 32);
        float recv2 = __shfl_xor(send2, 16, 32);

        // Branchless per-half selection: lower lanes get transforms 0-2,
        // upper lanes get transforms 3-5 (9 cndmasks, no exec churn).
        float t00 = upper ? d1[1] : d1[0];
        float t01 = upper ? d1[2] : d1[1];
        float t02 = upper ? d1[3] : d1[2];
        float t10 = upper ? d1[4] : d1[3];
        float t11 = upper ? d1[5] : d1[4];
        float t12 = upper ? d1[6] : d1[5];
        float t20 = upper ? d1[7] : d1[6];
        float t21 = upper ? recv1 : d1[7];
        float t22 = upper ? recv2 : recv1;

        probe(t00, t01, t02, acc0);
        probe(t10, t11, t12, acc1);
        probe(t20, t21, t22, acc2);
    }

    // --- Reduce: lower lanes carry transforms 0-2 (planes), upper 3-5 (rotations)
    int base = upper ? 3 : 0;
    atomicAdd(&accum[base + 0], acc0);
    atomicAdd(&accum[base + 1], acc1);
    atomicAdd(&accum[base + 2], acc2);
    __syncthreads();
    if (tid < 6) atomicAdd(&out[tid], accum[tid] * (1.0f / (float)B_SZ));
}

extern "C" void kernel_launch(void* const* d_in, const int* in_sizes, int n_in,
                              void* d_out, int out_size, void* d_ws, size_t ws_size,
                              hipStream_t stream) {
    (void)in_sizes; (void)n_in; (void)out_size; (void)d_ws; (void)ws_size;
    const float* planes  = (const float*)d_in[0];   // (3, 256, 4)
    const float* axes    = (const float*)d_in[1];   // (3, 256, 4)
    const float* samples = (const float*)d_in[2];   // (256, 4096, 3)
    const float* closest = (const float*)d_in[3];   // (256, 32768, 3)
    const float* voxel   = (const float*)d_in[4];   // (256, 32768)
    // d_in[5] = grid_size scalar (==32, compile-time constant here)
    float* out = (float*)d_out;                     // [ref0..2, rot0..2]

    symmetry_init<<<1, 768, 0, stream>>>(axes, out);
    dim3 grid(B_SZ, SPLIT);
    symmetry_main<<<grid, THREADS, 0, stream>>>(planes, axes, samples, closest, voxel, out);
}